// Calculator_31026843746318
// MI455X (gfx1250) — compile-verified
//
#include <hip/hip_runtime.h>

typedef unsigned int u32;
typedef __attribute__((ext_vector_type(4))) u32 v4u;
typedef __attribute__((ext_vector_type(8))) int v8i;
typedef __attribute__((ext_vector_type(4))) int v4i;

#define CHUNK 128            // edges per wave-chunk (4 per lane, wave32)
#define WAVES_PER_BLOCK 8    // 256 threads
#define BLOCKS 1024

// ---- TDM: 1-D dword copy global -> LDS (descriptor per CDNA5 ISA) ----
__device__ __forceinline__ void tdm_load_1d(u32 lds_off, const void* gptr, u32 ndwords) {
    unsigned long long ga = (unsigned long long)(size_t)gptr;
    v4u g0;
    g0[0] = 1u;                                   // count=1 (valid descriptor), user mode
    g0[1] = lds_off;                              // LDS byte address
    g0[2] = (u32)(ga & 0xFFFFFFFFu);              // global_addr[31:0]
    g0[3] = (u32)((ga >> 32) & 0x1FFFFFFu)        // global_addr[56:32]
          | (2u << 30);                           // type=2 ("image")
    v8i g1;
    g1[0] = (int)(2u << 16);                      // workgroup_mask=0, data_size=2 (4B)
    g1[1] = (int)((ndwords & 0xFFFFu) << 16);     // tensor_dim0[15:0] @ bits[63:48]
    g1[2] = (int)(((ndwords >> 16) & 0xFFFFu)     // tensor_dim0[31:16] @ bits[79:64]
          | (1u << 16));                          // tensor_dim1 = 1
    g1[3] = (int)((ndwords & 0xFFFFu) << 16);     // tile_dim0 = ndwords @ bits[127:112]
    g1[4] = 0;                                    // tile_dim1=0, tile_dim2=0 (unused)
    g1[5] = (int)ndwords;                         // tensor_dim0_stride[31:0]
    g1[6] = 0;
    g1[7] = 0;
    v4i z4 = {0, 0, 0, 0};                        // groups 2/3 unused (<=2D tensor)
    v8i z8 = {0, 0, 0, 0, 0, 0, 0, 0};            // 6-arg toolchain: extra group
    __builtin_amdgcn_tensor_load_to_lds(g0, g1, z4, z4, z8, 0);
}

__global__ __launch_bounds__(256) void edge_scatter_kernel(
    const float* __restrict__ charges,     // [N_ATOMS,4]
    const int*   __restrict__ nidx,        // [N_EDGES,2]
    const float* __restrict__ ndist,       // [N_EDGES]
    float*       __restrict__ out,         // [N_ATOMS,4]
    int nChunks, int cpw)                  // cpw = ceil(nChunks / totalWaves), host-side
{
    __shared__ int   sIdx [2][WAVES_PER_BLOCK][2 * CHUNK];  // 16 KB
    __shared__ float sDist[2][WAVES_PER_BLOCK][CHUNK];      //  8 KB

    const int lane = threadIdx.x & 31;
    const int wave = threadIdx.x >> 5;

    const int waveGlobal = blockIdx.x * WAVES_PER_BLOCK + wave;
    // contiguous chunk span per wave -> sequential TDM streaming
    const int c0 = waveGlobal * cpw;
    const int c1 = (c0 + cpw < nChunks) ? (c0 + cpw) : nChunks;
    if (c0 >= c1) return;   // per-wave exit; no cross-wave barriers used

    u32 ldsIdx0  = (u32)(size_t)&sIdx [0][wave][0];
    u32 ldsIdx1  = (u32)(size_t)&sIdx [1][wave][0];
    u32 ldsDist0 = (u32)(size_t)&sDist[0][wave][0];
    u32 ldsDist1 = (u32)(size_t)&sDist[1][wave][0];

    auto issue = [&](int buf, int chunk) {
        const int*   gi = nidx  + (size_t)chunk * 2 * CHUNK;
        const float* gd = ndist + (size_t)chunk * CHUNK;
        tdm_load_1d(buf ? ldsIdx1  : ldsIdx0,  gi, 2 * CHUNK);
        tdm_load_1d(buf ? ldsDist1 : ldsDist0, gd, CHUNK);
    };

    issue(0, c0);
    int buf = 0;
    for (int c = c0; c < c1; ++c, buf ^= 1) {
        if (c + 1 < c1) {
            issue(buf ^ 1, c + 1);                     // prefetch next chunk
            __builtin_amdgcn_s_wait_tensorcnt(2);      // current chunk's 2 TDM ops done
        } else {
            __builtin_amdgcn_s_wait_tensorcnt(0);
        }

        const int2*  pij = (const int2*)&sIdx[buf][wave][0];
        const float* pd  = &sDist[buf][wave][0];
#pragma unroll
        for (int k = 0; k < CHUNK / 32; ++k) {
            const int e  = k * 32 + lane;
            const int2  ij = pij[e];                   // ds_load_b64
            const float d  = pd[e];                    // ds_load_b32
            const float p  = 0.5f * __builtin_amdgcn_rcpf(d);  // PREFACTOR*0.5/d

            const float4 cj = *(const float4*)(charges + (size_t)ij.y * 4); // b128 gather
            const float4 ci = *(const float4*)(charges + (size_t)ij.x * 4);

            float* oi = out + (size_t)ij.x * 4;
            float* oj = out + (size_t)ij.y * 4;
            unsafeAtomicAdd(oi + 0, cj.x * p);
            unsafeAtomicAdd(oi + 1, cj.y * p);
            unsafeAtomicAdd(oi + 2, cj.z * p);
            unsafeAtomicAdd(oi + 3, cj.w * p);
            unsafeAtomicAdd(oj + 0, ci.x * p);
            unsafeAtomicAdd(oj + 1, ci.y * p);
            unsafeAtomicAdd(oj + 2, ci.z * p);
            unsafeAtomicAdd(oj + 3, ci.w * p);
        }
    }
}

__global__ void zero_out_kernel(float4* __restrict__ out, int n4) {
    int i = blockIdx.x * blockDim.x + threadIdx.x;
    if (i < n4) out[i] = make_float4(0.f, 0.f, 0.f, 0.f);
}

// Tail (edge count not divisible by CHUNK) — plain path, launched only if needed.
__global__ void edge_tail_kernel(
    const float* __restrict__ charges, const int* __restrict__ nidx,
    const float* __restrict__ ndist, float* __restrict__ out,
    int start, int nEdges)
{
    for (int e = start + blockIdx.x * blockDim.x + threadIdx.x; e < nEdges;
         e += gridDim.x * blockDim.x) {
        const int i = nidx[2 * e + 0];
        const int j = nidx[2 * e + 1];
        const float p = 0.5f * __builtin_amdgcn_rcpf(ndist[e]);
        const float4 cj = *(const float4*)(charges + (size_t)j * 4);
        const float4 ci = *(const float4*)(charges + (size_t)i * 4);
        float* oi = out + (size_t)i * 4;
        float* oj = out + (size_t)j * 4;
        unsafeAtomicAdd(oi + 0, cj.x * p); unsafeAtomicAdd(oi + 1, cj.y * p);
        unsafeAtomicAdd(oi + 2, cj.z * p); unsafeAtomicAdd(oi + 3, cj.w * p);
        unsafeAtomicAdd(oj + 0, ci.x * p); unsafeAtomicAdd(oj + 1, ci.y * p);
        unsafeAtomicAdd(oj + 2, ci.z * p); unsafeAtomicAdd(oj + 3, ci.w * p);
    }
}

extern "C" void kernel_launch(void* const* d_in, const int* in_sizes, int n_in,
                              void* d_out, int out_size, void* d_ws, size_t ws_size,
                              hipStream_t stream) {
    const float* charges = (const float*)d_in[0];
    // d_in[1] = cell (unused), d_in[2] = positions (unused)
    const int*   nidx    = (const int*)d_in[3];
    const float* ndist   = (const float*)d_in[4];
    float*       out     = (float*)d_out;

    const int nEdges = in_sizes[4];
    const int n4     = out_size / 4;        // float4 rows

    // out is poisoned by harness — zero it first (atomics accumulate into it)
    zero_out_kernel<<<(n4 + 255) / 256, 256, 0, stream>>>((float4*)out, n4);

    const int nChunks = nEdges / CHUNK;
    if (nChunks > 0) {
        const int totalWaves = BLOCKS * WAVES_PER_BLOCK;
        const int cpw = (nChunks + totalWaves - 1) / totalWaves;
        edge_scatter_kernel<<<BLOCKS, WAVES_PER_BLOCK * 32, 0, stream>>>(
            charges, nidx, ndist, out, nChunks, cpw);
    }
    const int tailStart = nChunks * CHUNK;
    if (tailStart < nEdges) {
        edge_tail_kernel<<<128, 256, 0, stream>>>(charges, nidx, ndist, out,
                                                  tailStart, nEdges);
    }
}